// PistonSpring_38079180046972
// MI455X (gfx1250) — compile-verified
//
#include <hip/hip_runtime.h>
#include <hip/hip_bf16.h>

// ---------------------------------------------------------------------------
// PistonSpring recurrence, CDNA5 (gfx1250), wave32.
// Layout: lane = state feature j (0..7), 4 batch elements per wave,
//         1024 waves total (128 blocks x 256 threads).
// Cross-lane: DPP8 fused into v_fmac/v_mul/v_add (VALU latency, no LDS).
// u staging: Tensor Data Mover -> LDS double buffer, TENSORcnt synced.
// ---------------------------------------------------------------------------

#if defined(__has_builtin)
#if __has_builtin(__builtin_amdgcn_tensor_load_to_lds) && \
    __has_builtin(__builtin_amdgcn_s_wait_tensorcnt)
#define USE_TDM 1
#endif
#endif
#ifndef USE_TDM
#define USE_TDM 0
#endif

typedef unsigned int __attribute__((ext_vector_type(4))) u32x4;
typedef int __attribute__((ext_vector_type(4))) i32x4;
typedef int __attribute__((ext_vector_type(8))) i32x8;

// ---- DPP8 fused VALU helpers (broadcast within each 8-lane group) ---------
#define DPP8_ALL(I) \
  "dpp8:[" #I "," #I "," #I "," #I "," #I "," #I "," #I "," #I "]"
// acc += bcast8(src, I) * coef
#define FMAC_DPP8(acc, src, coef, I)                         \
  asm("v_fmac_f32_dpp %0, %1, %2 " DPP8_ALL(I)               \
      : "+v"(acc)                                            \
      : "v"(src), "v"(coef))
// dst = bcast8(src, I) * coef
#define MUL_DPP8(dst, src, coef, I)                          \
  asm("v_mul_f32_dpp %0, %1, %2 " DPP8_ALL(I)                \
      : "=v"(dst)                                            \
      : "v"(src), "v"(coef))

// Sum across the 8 lanes of each group (XOR butterflies, fused adds);
// every lane ends with the total, identical bit pattern in all 8 lanes.
__device__ __forceinline__ float reduce8(float v) {
  float a, b, c;
  asm("v_add_f32_dpp %0, %1, %1 dpp8:[1,0,3,2,5,4,7,6]" : "=v"(a) : "v"(v));
  asm("v_add_f32_dpp %0, %1, %1 dpp8:[2,3,0,1,6,7,4,5]" : "=v"(b) : "v"(a));
  asm("v_add_f32_dpp %0, %1, %1 dpp8:[4,5,6,7,0,1,2,3]" : "=v"(c) : "v"(b));
  return c;
}

#if USE_TDM
// Issue a TDM load of a 32-row x 64-col f32 tile of u into LDS.
// gaddr points at u[batch_base][t0]; row stride = 1024 elements.
// D# bitfields per CDNA5 ISA ch.8 (group0: control/addr, group1: dims/tile).
__device__ __forceinline__ void tdm_load_u(const float* gaddr,
                                           unsigned lds_byte) {
  unsigned long long ga = (unsigned long long)gaddr;
  u32x4 g0;
  g0[0] = 1u;                                           // count=1, user mode
  g0[1] = lds_byte;                                     // lds_addr
  g0[2] = (unsigned)ga;                                 // global_addr[31:0]
  g0[3] = ((unsigned)(ga >> 32) & 0x1FFFFFFu) | (2u << 30);  // addr hi | type=2
  i32x8 g1;
  g1[0] = 2 << 16;            // workgroup_mask=0, data_size=2 (4 bytes)
  g1[1] = 1024 << 16;         // tensor_dim0[15:0]=1024 (time extent)
  g1[2] = 4096 << 16;         // tensor_dim0 hi=0 | tensor_dim1[15:0]=4096
  g1[3] = 64 << 16;           // tensor_dim1 hi=0 | tile_dim0=64
  g1[4] = 32;                 // tile_dim1=32 rows, tile_dim2=0
  g1[5] = 1024;               // tensor_dim0_stride[31:0]=1024 elements
  g1[6] = 0;                  // stride hi | tensor_dim1_stride lo
  g1[7] = 0;
  i32x4 g2 = {0, 0, 0, 0}, g3 = {0, 0, 0, 0};  // 2D tensor: groups 2/3 unused
  i32x8 g4 = {0, 0, 0, 0, 0, 0, 0, 0};         // extra group (6-arg builtin)
  __builtin_amdgcn_tensor_load_to_lds(g0, g1, g2, g3, g4, 0);
}
#else
// Cooperative fallback staging: each thread copies one 8-float row segment.
__device__ __forceinline__ void stage_u(float* dst, const float* u,
                                        int batch_base, int c, int tid) {
  const int idx = tid * 8;
  const int row = idx >> 6, col = idx & 63;
  const float* src = u + (size_t)(batch_base + row) * 1024 + c * 64 + col;
  float4 A = *(const float4*)src;
  float4 B = *(const float4*)(src + 4);
  *(float4*)(dst + row * 64 + col) = A;
  *(float4*)(dst + row * 64 + col + 4) = B;
}
#endif

// One recurrence step. ST=1: store x_{t+1} at op[ii*8] (immediate offset).
#define STEP(ii, ST)                                                          \
  {                                                                           \
    /* p_j = b_j + sum_i K[j][i] * x_i   (two fused accumulator chains) */    \
    float aA = bj, aB;                                                        \
    FMAC_DPP8(aA, x, ka[0], 0);                                               \
    MUL_DPP8(aB, x, ka[1], 1);                                                \
    FMAC_DPP8(aA, x, ka[2], 2);                                               \
    FMAC_DPP8(aB, x, ka[3], 3);                                               \
    FMAC_DPP8(aA, x, ka[4], 4);                                               \
    FMAC_DPP8(aB, x, ka[5], 5);                                               \
    FMAC_DPP8(aA, x, ka[6], 6);                                               \
    FMAC_DPP8(aB, x, ka[7], 7);                                               \
    const float pre = aA + aB;                                                \
    /* tanh(p) = 1 - 2/(exp(2p)+1), clamped (saturated) at |p|=9 */           \
    const float pc = __builtin_amdgcn_fmed3f(pre, -9.0f, 9.0f);               \
    const float e2 = __builtin_amdgcn_exp2f(pc * 2.88539008177793f);          \
    const float th = fmaf(-2.0f, __builtin_amdgcn_rcpf(e2 + 1.0f), 1.0f);     \
    /* dh_j = sum_i t_i * K[i][j] */                                          \
    float dA, dB;                                                             \
    MUL_DPP8(dA, th, kb[0], 0);                                               \
    MUL_DPP8(dB, th, kb[1], 1);                                               \
    FMAC_DPP8(dA, th, kb[2], 2);                                              \
    FMAC_DPP8(dB, th, kb[3], 3);                                              \
    FMAC_DPP8(dA, th, kb[4], 4);                                              \
    FMAC_DPP8(dB, th, kb[5], 5);                                              \
    FMAC_DPP8(dA, th, kb[6], 6);                                              \
    FMAC_DPP8(dB, th, kb[7], 7);                                              \
    const float dh = dA + dB;                                                 \
    /* gate = sigmoid( Wg . [x, dh] + bg ) : per-lane partial + butterfly */  \
    const float part = fmaf(wgx, x, fmaf(wgd, dh, bg8));                      \
    const float gs = reduce8(part);                                           \
    const float gate = __builtin_amdgcn_rcpf(                                 \
        1.0f + __builtin_amdgcn_exp2f(-1.44269504088896f * gs));              \
    float gamma;                                                              \
    MUL_DPP8(gamma, dh, gate, 3); /* gamma = gate * dh3 */                    \
    float lastsum = reduce8(tv * dh); /* sum T[i-1]*dh_i, i=1..6 */           \
    const float ngamma = -gamma;                                              \
    FMAC_DPP8(lastsum, dh, ngamma, 0); /* lastsum -= gamma*dh0 */             \
    float r, v0;                                                              \
    MUL_DPP8(r, dh, tm, 7);      /* mid rows 1..6: -T[j-1]*dh7 */             \
    MUL_DPP8(v0, dh, gamma, 7);  /* row 0: gamma*dh7 */                       \
    r = (j == 0) ? v0 : r;                                                    \
    r = (j == 7) ? lastsum : r;                                               \
    FMAC_DPP8(r, ur, gj, ii);    /* + G[j]*u_t (only j==7) */                 \
    x = fmaf(0.01f, r, x);                                                    \
    if (ST) op[(ii)*8] = x;                                                   \
  }

#define SUBCHUNK_FULL                                          \
  {                                                            \
    const float ur = *lp;                                      \
    lp += 8;                                                   \
    STEP(0, 1) STEP(1, 1) STEP(2, 1) STEP(3, 1)                \
    STEP(4, 1) STEP(5, 1) STEP(6, 1) STEP(7, 1)                \
    op += 64;                                                  \
  }

__global__ __launch_bounds__(256) void piston_spring_kernel(
    const float* __restrict__ x0, const float* __restrict__ u,
    const float* __restrict__ K, const float* __restrict__ b,
    const float* __restrict__ Tp, const float* __restrict__ G,
    const float* __restrict__ Wg, const float* __restrict__ bg,
    float* __restrict__ out) {
  __shared__ float ubuf[2][32 * 64];  // double-buffered u tiles (2 x 8KB)

  const int tid = threadIdx.x;
  const int j = tid & 7;        // state feature (DPP8 groups align with j)
  const int bl = tid >> 3;      // local batch 0..31
  const int batch_base = blockIdx.x * 32;
  const int batch = batch_base + bl;

  // Per-lane weight columns (all constant across the time loop).
  float ka[8], kb[8];
#pragma unroll
  for (int i = 0; i < 8; ++i) {
    ka[i] = K[j * 8 + i];  // K[j][i]  (for x @ K^T)
    kb[i] = K[i * 8 + j];  // K[i][j]  (for t @ K)
  }
  const float bj = b[j];
  const float tv = (j >= 1 && j <= 6) ? Tp[j - 1] : 0.0f;
  const float tm = -tv;
  const float gj = G[j];          // G[:,0]; only G[7] = -1
  const float wgx = Wg[j];
  const float wgd = Wg[8 + j];
  const float bg8 = bg[0] * 0.125f;  // bg split across 8 lanes of the sum

  float x = x0[batch * 8 + j];
  float* outp = out + (size_t)batch * 8192 + j;  // (batch*1024 + t)*8 + j
  *outp = x;                                     // preds[:, 0, :] = x0
  float* op = outp + 8;                          // next store: t = 1

#if USE_TDM
  if (tid < 32)  // wave 0 drives the Tensor Data Mover
    tdm_load_u(u + (size_t)batch_base * 1024,
               (unsigned)(unsigned long long)&ubuf[0][0]);
#endif

  // Chunks 0..14: 64 unguarded steps each (960 steps total).
#pragma unroll 1
  for (int c = 0; c < 15; ++c) {
#if USE_TDM
    if (tid < 32) __builtin_amdgcn_s_wait_tensorcnt(0);
    __syncthreads();  // publish tile c; all waves done reading buf (c&1)^1
    if (tid < 32)
      tdm_load_u(u + (size_t)batch_base * 1024 + (size_t)(c + 1) * 64,
                 (unsigned)(unsigned long long)&ubuf[(c & 1) ^ 1][0]);
#else
    __syncthreads();
    stage_u(&ubuf[c & 1][0], u, batch_base, c, tid);
    __syncthreads();
#endif
    const float* lp = &ubuf[c & 1][bl * 64 + j];
#pragma unroll 1
    for (int s8 = 0; s8 < 8; ++s8) SUBCHUNK_FULL
  }

  // Chunk 15: t = 960..1022 (63 steps; step t=1023 is never needed).
#if USE_TDM
  if (tid < 32) __builtin_amdgcn_s_wait_tensorcnt(0);
  __syncthreads();
#else
  __syncthreads();
  stage_u(&ubuf[1][0], u, batch_base, 15, tid);
  __syncthreads();
#endif
  {
    const float* lp = &ubuf[1][bl * 64 + j];
#pragma unroll 1
    for (int s8 = 0; s8 < 7; ++s8) SUBCHUNK_FULL
    // final subchunk: 7 steps (t = 1016..1022), no step for t = 1023
    const float ur = *lp;
    STEP(0, 1) STEP(1, 1) STEP(2, 1) STEP(3, 1)
    STEP(4, 1) STEP(5, 1) STEP(6, 1)
  }
}

extern "C" void kernel_launch(void* const* d_in, const int* in_sizes, int n_in,
                              void* d_out, int out_size, void* d_ws,
                              size_t ws_size, hipStream_t stream) {
  (void)in_sizes; (void)n_in; (void)out_size; (void)d_ws; (void)ws_size;
  const float* x0 = (const float*)d_in[0];
  const float* u = (const float*)d_in[1];
  const float* K = (const float*)d_in[2];
  const float* b = (const float*)d_in[3];
  const float* Tp = (const float*)d_in[4];
  const float* G = (const float*)d_in[5];
  const float* Wg = (const float*)d_in[6];
  const float* bg = (const float*)d_in[7];
  float* out = (float*)d_out;
  hipLaunchKernelGGL(piston_spring_kernel, dim3(128), dim3(256), 0, stream,
                     x0, u, K, b, Tp, G, Wg, bg, out);
}